// Gat_38465727103402
// MI455X (gfx1250) — compile-verified
//
#include <hip/hip_runtime.h>
#include <hip/hip_bf16.h>

// Only meaningful to check in the device pass; host __has_builtin can't see
// amdgcn target builtins (round-1 feedback: device pass was clean, host errored).
#if defined(__HIP_DEVICE_COMPILE__) && !__has_builtin(__builtin_amdgcn_wmma_f32_16x16x4_f32)
#error "missing __builtin_amdgcn_wmma_f32_16x16x4_f32 on this toolchain (device pass)"
#endif

typedef float v2f __attribute__((ext_vector_type(2)));
typedef float v8f __attribute__((ext_vector_type(8)));

#define D_DIM 256
#define N_NODES 20000
#define E_EDGES 640000
#define G_GRAPHS 64
#define EN_TOT (E_EDGES + N_NODES)

// ---------------------------------------------------------------------------
// GEMM: H[M x 256] = X[M x 256] @ W[256 x 256], fp32 WMMA 16x16x4.
// One wave32 per 16x16 output tile. M and 256 are multiples of 16, so the
// branch below is wave-uniform (EXEC stays all-ones across every v_wmma).
// ---------------------------------------------------------------------------
__global__ __launch_bounds__(256) void gemm_f32_wmma(const float* __restrict__ Xm,
                                                     const float* __restrict__ Wm,
                                                     float* __restrict__ Hm,
                                                     int Mrows) {
  const int wave = (blockIdx.x * blockDim.x + threadIdx.x) >> 5;
  const int lane = threadIdx.x & 31;
  const int ntiles = D_DIM / 16;                    // 16 tiles along N
  const int mt = wave / ntiles;
  const int nt = wave % ntiles;
  if (mt * 16 >= Mrows) return;                     // wave-uniform
  const int m0 = mt * 16, n0 = nt * 16;
  const int half = lane >> 4;                       // 0 or 1
  const int l = lane & 15;

  v8f c = {0.f, 0.f, 0.f, 0.f, 0.f, 0.f, 0.f, 0.f};
  // A (16x4 f32): lane l, half h -> A[m0+l][k + 2h + {0,1}]
  const float* arow = Xm + (size_t)(m0 + l) * D_DIM + 2 * half;
  // B (4x16 f32): lane l, half h -> B[k + 2h + {0,1}][n0+l]
  const float* bcol = Wm + (size_t)(2 * half) * D_DIM + (n0 + l);

  for (int k = 0; k < D_DIM; k += 4) {
    v2f a, b;
    a.x = arow[k];
    a.y = arow[k + 1];
    const float* bk = bcol + (size_t)k * D_DIM;
    b.x = bk[0];
    b.y = bk[D_DIM];
    c = __builtin_amdgcn_wmma_f32_16x16x4_f32(false, a, false, b,
                                              (short)0, c, false, false);
  }

  // C/D layout: VGPR r, lanes 0-15 -> M=m0+r, lanes 16-31 -> M=m0+8+r, N=n0+l
  float* hp = Hm + (size_t)(m0 + half * 8) * D_DIM + (n0 + l);
#pragma unroll
  for (int r = 0; r < 8; ++r) hp[(size_t)r * D_DIM] = c[r];
}

// ---------------------------------------------------------------------------
// alpha_s = H @ a_src, alpha_d = H @ a_dst : one wave32 per node row.
// ---------------------------------------------------------------------------
__global__ __launch_bounds__(256) void alpha_kernel(const float* __restrict__ H,
                                                    const float* __restrict__ asrc,
                                                    const float* __restrict__ adst,
                                                    float* __restrict__ als,
                                                    float* __restrict__ ald, int n) {
  const int wave = (blockIdx.x * blockDim.x + threadIdx.x) >> 5;
  const int lane = threadIdx.x & 31;
  if (wave >= n) return;
  const float* hr = H + (size_t)wave * D_DIM;
  float ss = 0.f, sd = 0.f;
#pragma unroll
  for (int j = 0; j < D_DIM / 32; ++j) {
    const int d = j * 32 + lane;
    const float hv = hr[d];
    ss = fmaf(hv, asrc[d], ss);
    sd = fmaf(hv, adst[d], sd);
  }
#pragma unroll
  for (int off = 16; off > 0; off >>= 1) {
    ss += __shfl_xor(ss, off, 32);
    sd += __shfl_xor(sd, off, 32);
  }
  if (lane == 0) { als[wave] = ss; ald[wave] = sd; }
}

// Order-preserving float<->uint mapping for atomicMax on floats.
__device__ __forceinline__ unsigned fflip(float f) {
  unsigned u = __float_as_uint(f);
  return u ^ (unsigned)(((int)u >> 31) | 0x80000000);
}
__device__ __forceinline__ float funflip(unsigned u) {
  unsigned m = (u >> 31) ? 0x80000000u : 0xFFFFFFFFu;
  return __uint_as_float(u ^ m);
}

__device__ __forceinline__ void edge_ends(const int* __restrict__ ei, int e,
                                          int& s, int& d) {
  if (e < E_EDGES) { s = ei[e]; d = ei[E_EDGES + e]; }
  else             { s = d = e - E_EDGES; }          // appended self-loops
}

// Pass A: score = leaky_relu(as[src] + ad[dst], 0.2); segment max over dst.
__global__ void edge_score_max(const int* __restrict__ ei,
                               const float* __restrict__ als,
                               const float* __restrict__ ald,
                               float* __restrict__ scores,
                               unsigned* __restrict__ mmax) {
  const int e = blockIdx.x * blockDim.x + threadIdx.x;
  if (e >= EN_TOT) return;
  int s, d; edge_ends(ei, e, s, d);
  float sc = als[s] + ald[d];
  sc = sc > 0.f ? sc : 0.2f * sc;
  scores[e] = sc;
  atomicMax(&mmax[d], fflip(sc));
}

// Pass B: ex = exp(score - max[dst]); segment sum over dst.
__global__ void edge_exp_sum(const int* __restrict__ ei,
                             const unsigned* __restrict__ mmax,
                             float* __restrict__ scores,
                             float* __restrict__ ssum) {
  const int e = blockIdx.x * blockDim.x + threadIdx.x;
  if (e >= EN_TOT) return;
  int s, d; edge_ends(ei, e, s, d);
  const float ex = __expf(scores[e] - funflip(mmax[d]));
  scores[e] = ex;
  atomicAdd(&ssum[d], ex);
}

// Pass C: out[dst] += (ex/sum[dst]) * h[src]  — one wave per edge,
// 8 coalesced lane-strided f32 atomic adds per lane. Dominant memory phase.
__global__ __launch_bounds__(256) void edge_aggregate(const int* __restrict__ ei,
                                                      const float* __restrict__ H,
                                                      const float* __restrict__ scores,
                                                      const float* __restrict__ ssum,
                                                      float* __restrict__ out) {
  const int e = (blockIdx.x * blockDim.x + threadIdx.x) >> 5;
  const int lane = threadIdx.x & 31;
  if (e >= EN_TOT) return;
  int s, d; edge_ends(ei, e, s, d);
  const float w = scores[e] / ssum[d];
  const float* hs = H + (size_t)s * D_DIM;
  float* od = out + (size_t)d * D_DIM;
#pragma unroll
  for (int j = 0; j < D_DIM / 32; ++j) {
    const int c = j * 32 + lane;
    atomicAdd(&od[c], w * hs[c]);
  }
}

__global__ void bias_lrelu(float* __restrict__ h, const float* __restrict__ b,
                           int total) {
  const int i = blockIdx.x * blockDim.x + threadIdx.x;
  if (i >= total) return;
  float v = h[i] + b[i & (D_DIM - 1)];
  h[i] = v > 0.f ? v : 0.01f * v;
}

__global__ void count_nodes(const int* __restrict__ batch, float* __restrict__ cnt,
                            int n) {
  const int i = blockIdx.x * blockDim.x + threadIdx.x;
  if (i < n) atomicAdd(&cnt[batch[i]], 1.0f);
}

// BatchNorm (eval) fused with segment-sum pooling into pooled[G x D].
__global__ void bn_pool(const float* __restrict__ h, const int* __restrict__ batch,
                        const float* __restrict__ gam, const float* __restrict__ bet,
                        const float* __restrict__ mu, const float* __restrict__ var,
                        float* __restrict__ pooled, int n) {
  const int i = blockIdx.x * blockDim.x + threadIdx.x;
  if (i >= n * D_DIM) return;
  const int node = i >> 8, d = i & (D_DIM - 1);
  const float y = (h[i] - mu[d]) * rsqrtf(var[d] + 1e-5f) * gam[d] + bet[d];
  atomicAdd(&pooled[(size_t)batch[node] * D_DIM + d], y);
}

// out[g] = (pooled[g]/max(cnt,1)) @ Wl^T + bl ; one block per graph.
__global__ __launch_bounds__(256) void final_linear(const float* __restrict__ pooled,
                                                    const float* __restrict__ cnt,
                                                    const float* __restrict__ Wl,
                                                    const float* __restrict__ bl,
                                                    float* __restrict__ out) {
  __shared__ float row[D_DIM];
  const int g = blockIdx.x, t = threadIdx.x;
  float c = cnt[g];
  c = c > 1.f ? c : 1.f;
  row[t] = pooled[(size_t)g * D_DIM + t] / c;
  __syncthreads();
  const float* wr = Wl + (size_t)t * D_DIM;   // Wl[t][k]
  float acc = 0.f;
#pragma unroll 4
  for (int k = 0; k < D_DIM; ++k) acc = fmaf(row[k], wr[k], acc);
  out[(size_t)g * D_DIM + t] = acc + bl[t];
}

// ---------------------------------------------------------------------------
static void run_gat_layer(const float* xin, const float* Wm, const float* asrc,
                          const float* adst, const float* bias, float* h,
                          float* xout, float* als, float* ald, unsigned* mmax,
                          float* ssum, float* scores, const int* ei,
                          hipStream_t stream) {
  const int gemm_waves = (N_NODES / 16) * (D_DIM / 16);        // 20000
  gemm_f32_wmma<<<gemm_waves / 8, 256, 0, stream>>>(xin, Wm, h, N_NODES);
  alpha_kernel<<<(N_NODES * 32 + 255) / 256, 256, 0, stream>>>(h, asrc, adst,
                                                               als, ald, N_NODES);
  (void)hipMemsetAsync(mmax, 0, N_NODES * sizeof(unsigned), stream);
  (void)hipMemsetAsync(ssum, 0, N_NODES * sizeof(float), stream);
  edge_score_max<<<(EN_TOT + 255) / 256, 256, 0, stream>>>(ei, als, ald, scores, mmax);
  edge_exp_sum<<<(EN_TOT + 255) / 256, 256, 0, stream>>>(ei, mmax, scores, ssum);
  (void)hipMemsetAsync(xout, 0, (size_t)N_NODES * D_DIM * sizeof(float), stream);
  edge_aggregate<<<((size_t)EN_TOT * 32 + 255) / 256, 256, 0, stream>>>(ei, h, scores,
                                                                        ssum, xout);
  bias_lrelu<<<(N_NODES * D_DIM + 255) / 256, 256, 0, stream>>>(xout, bias,
                                                                N_NODES * D_DIM);
}

extern "C" void kernel_launch(void* const* d_in, const int* in_sizes, int n_in,
                              void* d_out, int out_size, void* d_ws, size_t ws_size,
                              hipStream_t stream) {
  const float* x      = (const float*)d_in[0];
  const int*   ei     = (const int*)d_in[1];
  const int*   batch  = (const int*)d_in[2];
  const float* W1     = (const float*)d_in[3];
  const float* a1s    = (const float*)d_in[4];
  const float* a1d    = (const float*)d_in[5];
  const float* b1     = (const float*)d_in[6];
  const float* W2     = (const float*)d_in[7];
  const float* a2s    = (const float*)d_in[8];
  const float* a2d    = (const float*)d_in[9];
  const float* b2     = (const float*)d_in[10];
  const float* bng    = (const float*)d_in[11];
  const float* bnb    = (const float*)d_in[12];
  const float* bnm    = (const float*)d_in[13];
  const float* bnv    = (const float*)d_in[14];
  const float* Wl     = (const float*)d_in[15];
  const float* bl     = (const float*)d_in[16];

  char* ws = (char*)d_ws;
  const size_t ND = (size_t)N_NODES * D_DIM * sizeof(float);   // 20.48 MB
  size_t off = 0;
  float*    A      = (float*)(ws + off); off += ND;
  float*    B      = (float*)(ws + off); off += ND;
  float*    als    = (float*)(ws + off); off += N_NODES * sizeof(float);
  float*    ald    = (float*)(ws + off); off += N_NODES * sizeof(float);
  unsigned* mmax   = (unsigned*)(ws + off); off += N_NODES * sizeof(unsigned);
  float*    ssum   = (float*)(ws + off); off += N_NODES * sizeof(float);
  float*    scores = (float*)(ws + off); off += (size_t)EN_TOT * sizeof(float);
  float*    pooled = (float*)(ws + off); off += (size_t)G_GRAPHS * D_DIM * sizeof(float);
  float*    cnt    = (float*)(ws + off); off += G_GRAPHS * sizeof(float);
  if (off > ws_size) return;   // workspace too small: bail (nothing launched)

  // Layer 1: x -> B
  run_gat_layer(x, W1, a1s, a1d, b1, A, B, als, ald, mmax, ssum, scores, ei, stream);
  // Layer 2: B -> B (h staged in A; B is dead after the GEMM reads it)
  run_gat_layer(B, W2, a2s, a2d, b2, A, B, als, ald, mmax, ssum, scores, ei, stream);

  // BN + mean pool + final linear
  (void)hipMemsetAsync(pooled, 0, (size_t)G_GRAPHS * D_DIM * sizeof(float), stream);
  (void)hipMemsetAsync(cnt, 0, G_GRAPHS * sizeof(float), stream);
  count_nodes<<<(N_NODES + 255) / 256, 256, 0, stream>>>(batch, cnt, N_NODES);
  bn_pool<<<(N_NODES * D_DIM + 255) / 256, 256, 0, stream>>>(B, batch, bng, bnb,
                                                             bnm, bnv, pooled, N_NODES);
  final_linear<<<G_GRAPHS, 256, 0, stream>>>(pooled, cnt, Wl, bl, (float*)d_out);
}